// Network_5772436046487
// MI455X (gfx1250) — compile-verified
//
#include <hip/hip_runtime.h>

// Problem constants (match reference)
#define N_NODES 50000
#define N_EDGES 1600000
#define T_STEPS 50
#define BATCH   8
#define DT_STEP 0.02f

#define TILE 256                      // edges per tile == blockDim
#define SCATTER_BLOCKS 1024           // persistent-ish blocks looping over tiles

// ---------------------------------------------------------------------------
// gfx1250 async-LDS path detection (compile-robust across toolchains)
// ---------------------------------------------------------------------------
#if defined(__AMDGCN__) && __has_builtin(__builtin_amdgcn_global_load_async_to_lds_b128)
#define USE_ASYNC_LDS 1
#else
#define USE_ASYNC_LDS 0
#endif

#if USE_ASYNC_LDS
#if __has_builtin(__builtin_amdgcn_s_wait_asynccnt)
#define WAIT_ASYNC(n)                                  \
    do {                                               \
        __builtin_amdgcn_s_wait_asynccnt(n);           \
        asm volatile("" ::: "memory");                 \
    } while (0)
#else
#define WAIT_ASYNC(n)                                  \
    asm volatile("s_wait_asynccnt " #n ::: "memory")
#endif

// Builtin expects pointers to int __vector(4) in explicit address spaces:
//   arg0: global (AS1, printed as '__device__' by clang-HIP) v4i*
//   arg1: LDS    (AS3) v4i*
typedef int v4i __attribute__((ext_vector_type(4)));
typedef __attribute__((address_space(1))) v4i as1_v4i;
typedef __attribute__((address_space(3))) v4i as3_v4i;

// Issue a 16B async copy: global edge record -> this lane's LDS slot.
// AS1 ptr from the 64-bit flat address; AS3 ptr from the truncated 32-bit
// LDS offset (generic shared address low bits == LDS byte offset).
__device__ __forceinline__ void async_fetch_b128(const void* g, void* l) {
    __builtin_amdgcn_global_load_async_to_lds_b128(
        (as1_v4i*)(unsigned long long)(uintptr_t)g,
        (as3_v4i*)(unsigned int)(uintptr_t)l,
        0, 0);
    asm volatile("" ::: "memory");
}
#endif // USE_ASYNC_LDS

// ---------------------------------------------------------------------------
// Kernel 1: pack edges into 16B records (src, tgt, weight, pad) for single
// b128 loads in the hot loop. weight = sign * syn_count * max(strength, 0).
// ---------------------------------------------------------------------------
__global__ void build_edges_kernel(const float* __restrict__ sign,
                                   const float* __restrict__ syn_count,
                                   const float* __restrict__ syn_strength,
                                   const int*   __restrict__ source_idx,
                                   const int*   __restrict__ target_idx,
                                   int4* __restrict__ epack,
                                   int E) {
    int e = blockIdx.x * blockDim.x + threadIdx.x;
    if (e >= E) return;
    float w = sign[e] * syn_count[e] * fmaxf(syn_strength[e], 0.0f);
    int4 p;
    p.x = source_idx[e];
    p.y = target_idx[e];
    p.z = __float_as_int(w);
    p.w = 0;
    epack[e] = p;
}

// ---------------------------------------------------------------------------
// Kernel 2: per-node init. a = DT / max(tau, DT); v[n][b] = bias[n]; accum = 0.
// v / accum layout is (N, 8): batch contiguous per node -> float4-friendly.
// ---------------------------------------------------------------------------
__global__ void init_nodes_kernel(const float* __restrict__ bias,
                                  const float* __restrict__ time_const,
                                  float* __restrict__ a,
                                  float* __restrict__ v,
                                  float* __restrict__ accum,
                                  int N) {
    int n = blockIdx.x * blockDim.x + threadIdx.x;
    if (n >= N) return;
    a[n] = DT_STEP / fmaxf(time_const[n], DT_STEP);
    float b = bias[n];
    float4 bb = make_float4(b, b, b, b);
    float4 zz = make_float4(0.f, 0.f, 0.f, 0.f);
    float4* v4  = reinterpret_cast<float4*>(v)     + (size_t)n * 2;
    float4* ac4 = reinterpret_cast<float4*>(accum) + (size_t)n * 2;
    v4[0] = bb;  v4[1] = bb;
    ac4[0] = zz; ac4[1] = zz;
}

// ---------------------------------------------------------------------------
// Shared per-edge work: gather v[src] (two b128, L2-resident), relu, scale,
// 8x global_atomic_add_f32 into the contiguous target slot.
// ---------------------------------------------------------------------------
__device__ __forceinline__ void process_edge(const int4 p,
                                             const float* __restrict__ v,
                                             float* __restrict__ accum) {
    int   s = p.x;
    int   t = p.y;
    float w = __int_as_float(p.z);

    const float4* vv = reinterpret_cast<const float4*>(v + (size_t)s * BATCH);
    float4 v0 = vv[0];
    float4 v1 = vv[1];

    float* acc = accum + (size_t)t * BATCH;
    atomicAdd(acc + 0, w * fmaxf(v0.x, 0.0f));
    atomicAdd(acc + 1, w * fmaxf(v0.y, 0.0f));
    atomicAdd(acc + 2, w * fmaxf(v0.z, 0.0f));
    atomicAdd(acc + 3, w * fmaxf(v0.w, 0.0f));
    atomicAdd(acc + 4, w * fmaxf(v1.x, 0.0f));
    atomicAdd(acc + 5, w * fmaxf(v1.y, 0.0f));
    atomicAdd(acc + 6, w * fmaxf(v1.z, 0.0f));
    atomicAdd(acc + 7, w * fmaxf(v1.w, 0.0f));
}

// ---------------------------------------------------------------------------
// Kernel 3 (hot): tiles of 256 edge records are double-buffered into LDS via
// GLOBAL_LOAD_ASYNC_TO_LDS_B128 (ASYNCcnt-tracked) so the async engine streams
// tile k+1 while the wave does the gather+atomic work of tile k. Each lane
// consumes exactly the record it fetched -> per-wave s_wait_asynccnt, no
// barriers. Fallback (non-gfx1250 / older toolchain): direct b128 loads.
// ---------------------------------------------------------------------------
__global__ void scatter_edges_kernel(const int4* __restrict__ epack,
                                     const float* __restrict__ v,
                                     float* __restrict__ accum,
                                     int E, int nTiles) {
    const int tid = threadIdx.x;

#if USE_ASYNC_LDS
    __shared__ int4 ebuf[2][TILE];

    int tile = blockIdx.x;
    const int stride = gridDim.x;
    int buf = 0;

    if (tile < nTiles) {
        int e = tile * TILE + tid;
        if (e < E) async_fetch_b128(&epack[e], &ebuf[0][tid]);
    }

    for (; tile < nTiles; tile += stride) {
        int nextTile = tile + stride;
        if (nextTile < nTiles) {
            int en = nextTile * TILE + tid;
            if (en < E) async_fetch_b128(&epack[en], &ebuf[buf ^ 1][tid]);
            WAIT_ASYNC(1);   // current tile complete; next may stay in flight
        } else {
            WAIT_ASYNC(0);
        }

        int e = tile * TILE + tid;
        if (e < E) {
            int4 p = ebuf[buf][tid];   // ds_load_b128
            process_edge(p, v, accum);
        }
        buf ^= 1;
    }
#else
    int tile = blockIdx.x;
    const int stride = gridDim.x;
    for (; tile < nTiles; tile += stride) {
        int e = tile * TILE + tid;
        if (e < E) {
            int pe = e + 8192;
            if (pe < E) __builtin_prefetch(&epack[pe], 0, 0);
            int4 p = epack[e];
            process_edge(p, v, accum);
        }
    }
#endif
}

// ---------------------------------------------------------------------------
// Kernel 4: fused Euler update. One thread per node handles all 8 batches.
//   v_next = v + a*(bias + syn + x_t - v)
// Writes vs[t] (layout (B,N) within the timestep), updates v, zeroes accum.
// ---------------------------------------------------------------------------
__global__ void update_nodes_kernel(const float* __restrict__ x_t,   // x + t*B*N
                                    const float* __restrict__ bias,
                                    const float* __restrict__ a,
                                    float* __restrict__ v,
                                    float* __restrict__ accum,
                                    float* __restrict__ out_t,       // out + t*B*N
                                    int N) {
    int n = blockIdx.x * blockDim.x + threadIdx.x;
    if (n >= N) return;

    float an = a[n];
    float bn = bias[n];

    float4* v4  = reinterpret_cast<float4*>(v)     + (size_t)n * 2;
    float4* ac4 = reinterpret_cast<float4*>(accum) + (size_t)n * 2;

    float4 vl0 = v4[0];
    float4 vl1 = v4[1];
    float4 s0  = ac4[0];
    float4 s1  = ac4[1];

    float vb[BATCH] = {vl0.x, vl0.y, vl0.z, vl0.w, vl1.x, vl1.y, vl1.z, vl1.w};
    float sb[BATCH] = {s0.x,  s0.y,  s0.z,  s0.w,  s1.x,  s1.y,  s1.z,  s1.w};
    float nv[BATCH];

#pragma unroll
    for (int b = 0; b < BATCH; ++b) {
        float xv = x_t[(size_t)b * N + n];
        float vn = vb[b] + an * (bn + sb[b] + xv - vb[b]);
        out_t[(size_t)b * N + n] = vn;
        nv[b] = vn;
    }

    v4[0] = make_float4(nv[0], nv[1], nv[2], nv[3]);
    v4[1] = make_float4(nv[4], nv[5], nv[6], nv[7]);

    float4 zz = make_float4(0.f, 0.f, 0.f, 0.f);
    ac4[0] = zz;
    ac4[1] = zz;
}

// ---------------------------------------------------------------------------
// Host-side orchestration (graph-capture safe: only kernel launches on stream)
// ---------------------------------------------------------------------------
static inline size_t align256(size_t x) { return (x + 255) & ~(size_t)255; }

extern "C" void kernel_launch(void* const* d_in, const int* in_sizes, int n_in,
                              void* d_out, int out_size, void* d_ws, size_t ws_size,
                              hipStream_t stream) {
    const float* x            = (const float*)d_in[0]; // (T, B, N)
    const float* bias         = (const float*)d_in[1]; // (N,)
    const float* time_const   = (const float*)d_in[2]; // (N,)
    const float* sign         = (const float*)d_in[3]; // (E,)
    const float* syn_count    = (const float*)d_in[4]; // (E,)
    const float* syn_strength = (const float*)d_in[5]; // (E,)
    const int*   source_idx   = (const int*)d_in[6];   // (E,)
    const int*   target_idx   = (const int*)d_in[7];   // (E,)
    float*       out          = (float*)d_out;         // (T, B, N)

    const int N = N_NODES;
    const int E = N_EDGES;
    const int nTiles = (E + TILE - 1) / TILE;          // 6250 (exact: E % TILE == 0)

    // Workspace carve-up (all L2-resident working state): ~29 MB total.
    char* ws = (char*)d_ws;
    size_t off = 0;
    int4*  epack = (int4*)(ws + off);  off += align256((size_t)E * sizeof(int4));
    float* a     = (float*)(ws + off); off += align256((size_t)N * sizeof(float));
    float* v     = (float*)(ws + off); off += align256((size_t)N * BATCH * sizeof(float));
    float* accum = (float*)(ws + off); off += align256((size_t)N * BATCH * sizeof(float));
    (void)ws_size; (void)n_in; (void)in_sizes; (void)out_size;

    const int BLK = 256;
    const int gridE = (E + BLK - 1) / BLK;
    const int gridN = (N + BLK - 1) / BLK;
    const int gridS = (nTiles < SCATTER_BLOCKS) ? nTiles : SCATTER_BLOCKS;

    build_edges_kernel<<<gridE, BLK, 0, stream>>>(sign, syn_count, syn_strength,
                                                  source_idx, target_idx, epack, E);
    init_nodes_kernel<<<gridN, BLK, 0, stream>>>(bias, time_const, a, v, accum, N);

    for (int t = 0; t < T_STEPS; ++t) {
        const float* x_t   = x   + (size_t)t * BATCH * N;
        float*       out_t = out + (size_t)t * BATCH * N;
        scatter_edges_kernel<<<gridS, TILE, 0, stream>>>(epack, v, accum, E, nTiles);
        update_nodes_kernel<<<gridN, BLK, 0, stream>>>(x_t, bias, a, v, accum, out_t, N);
    }
}